// DYS_opt_net_55224689492455
// MI455X (gfx1250) — compile-verified
//
#include <hip/hip_runtime.h>

// ---------------------------------------------------------------------------
// DYS_opt_net on MI455X (gfx1250): f32 WMMA GEMM pipeline.
//   project_C2(u) = P u + c,  P = I - A^T M^-1 A,  M = A A^T (SPD, 512x512),
//   M^-1 via Newton-Schulz (20 iters of GEMMs), all heavy math through
//   V_WMMA_F32_16X16X4_F32. Device-side convergence flag emulates the
//   reference while-loop under graph capture (fixed launch sequence).
// ---------------------------------------------------------------------------

typedef float v2f __attribute__((ext_vector_type(2)));
typedef float v8f __attribute__((ext_vector_type(8)));

#define N1    512
#define N2    2048
#define DDIM  2048
#define BATCH 512
#define ALPHA 0.05f
#define EPSTOL 0.01f
#define NEWTON_ITERS 20
#define DYS_ITERS 50

struct Ctrl {
    int      flag;       // 1 = converged, iteration kernels no-op
    int      depth;
    float    diff2;      // accumulated ||z_new - z||^2
    unsigned norm_bits;  // atomicMax target for ||M||_1 (positive floats)
    int      zero;       // always-0 skip flag for setup GEMMs
};

// ---- WMMA f32 16x16x4 --------------------------------------------------------
__device__ __forceinline__ v8f wmma_f32(v2f a, v2f b, v8f c) {
    // (neg_a, A, neg_b, B, c_mod, C, reuse_a, reuse_b)
    return __builtin_amdgcn_wmma_f32_16x16x4_f32(false, a, false, b, (short)0, c,
                                                 false, false);
}

// A fragment (16x4): lane 0-15 -> M, VGPR0/1 hold K = 2*laneHalf, +1
template <bool TRANS>
__device__ __forceinline__ v2f load_fragA(const float* __restrict__ A, int ld,
                                          int m0, int k0, int li, int lh) {
    if constexpr (!TRANS) {  // A stored [M, ld] row-major
        const float* p = A + (size_t)(m0 + li) * ld + (k0 + 2 * lh);
        return v2f{p[0], p[1]};
    } else {                 // op A[m,k] = A[k*ld + m]
        const float* p = A + (size_t)(k0 + 2 * lh) * ld + (m0 + li);
        return v2f{p[0], p[ld]};
    }
}

// B fragment (4x16): lane 0-15 -> N, VGPR0/1 hold K = 2*laneHalf, +1
template <bool TRANS>
__device__ __forceinline__ v2f load_fragB(const float* __restrict__ B, int ld,
                                          int k0, int n0, int li, int lh) {
    if constexpr (!TRANS) {  // B stored [K, ld] row-major
        const float* p = B + (size_t)(k0 + 2 * lh) * ld + (n0 + li);
        return v2f{p[0], p[ld]};
    } else {                 // op B[k,n] = B[n*ld + k]
        const float* p = B + (size_t)(n0 + li) * ld + (k0 + 2 * lh);
        return v2f{p[0], p[1]};
    }
}

// C/D fragment: VGPR r, lane half lh -> m = m0 + r + 8*lh, n = n0 + li
// EPI: 0 = store, 1 = +bias[n], 2 = 2*aux[idx]-acc (Newton), 3 = (m==n)-acc
template <int EPI>
__device__ __forceinline__ void store_frag(float* __restrict__ C,
                                           const float* __restrict__ aux,
                                           int ldc, int m0, int n0,
                                           const v8f& acc, int li, int lh) {
#pragma unroll
    for (int r = 0; r < 8; ++r) {
        int m = m0 + r + 8 * lh;
        int n = n0 + li;
        size_t idx = (size_t)m * ldc + n;
        float v = acc[r];
        if constexpr (EPI == 1) v += aux[n];
        else if constexpr (EPI == 2) v = 2.0f * aux[idx] - v;
        else if constexpr (EPI == 3) v = (m == n ? 1.0f : 0.0f) - v;
        C[idx] = v;
    }
}

// ---- Tiled WMMA GEMM: C[M,N] = opA(A)[M,K] * opB(B)[K,N]  (+ epilogue) ------
// Block = 256 threads = 8 waves (2x4), wave tile 32x32 -> block tile 64x128.
template <bool TA, bool TB, int EPI>
__global__ __launch_bounds__(256) void k_gemm(const float* __restrict__ A,
                                              const float* __restrict__ B,
                                              const float* __restrict__ aux,
                                              float* __restrict__ C,
                                              int K, int lda, int ldb, int ldc,
                                              const int* __restrict__ skip) {
    if (*(volatile const int*)skip) return;  // uniform: EXEC stays all-1 for WMMA

    const int lane = threadIdx.x & 31;
    const int wv   = threadIdx.x >> 5;
    const int li   = lane & 15;
    const int lh   = lane >> 4;
    const int m0   = blockIdx.y * 64 + (wv >> 2) * 32;
    const int n0   = blockIdx.x * 128 + (wv & 3) * 32;

    v8f acc00 = {}, acc01 = {}, acc10 = {}, acc11 = {};

#pragma unroll 2
    for (int k = 0; k < K; k += 4) {
        v2f a0 = load_fragA<TA>(A, lda, m0,      k, li, lh);
        v2f a1 = load_fragA<TA>(A, lda, m0 + 16, k, li, lh);
        v2f b0 = load_fragB<TB>(B, ldb, k, n0,      li, lh);
        v2f b1 = load_fragB<TB>(B, ldb, k, n0 + 16, li, lh);
        acc00 = wmma_f32(a0, b0, acc00);
        acc01 = wmma_f32(a0, b1, acc01);
        acc10 = wmma_f32(a1, b0, acc10);
        acc11 = wmma_f32(a1, b1, acc11);
    }

    store_frag<EPI>(C, aux, ldc, m0,      n0,      acc00, li, lh);
    store_frag<EPI>(C, aux, ldc, m0,      n0 + 16, acc01, li, lh);
    store_frag<EPI>(C, aux, ldc, m0 + 16, n0,      acc10, li, lh);
    store_frag<EPI>(C, aux, ldc, m0 + 16, n0 + 16, acc11, li, lh);
}

// ---- Small kernels ----------------------------------------------------------
__global__ void k_init(Ctrl* c) {
    c->flag = 0; c->depth = 0; c->diff2 = 0.0f; c->norm_bits = 0u; c->zero = 0;
}

// ||M||_1 = max row abs-sum (M symmetric) -> ctrl->norm_bits via atomicMax
__global__ __launch_bounds__(256) void k_rownorm(const float* __restrict__ M, Ctrl* ctrl) {
    __shared__ float s[256];
    const int row = blockIdx.x;
    float sum = 0.0f;
    for (int j = threadIdx.x; j < N1; j += 256) sum += fabsf(M[(size_t)row * N1 + j]);
    s[threadIdx.x] = sum;
    __syncthreads();
    for (int st = 128; st > 0; st >>= 1) {
        if (threadIdx.x < st) s[threadIdx.x] += s[threadIdx.x + st];
        __syncthreads();
    }
    if (threadIdx.x == 0)
        atomicMax((unsigned*)&ctrl->norm_bits, __float_as_uint(s[0]));
}

// X0 = M / ||M||_1^2  (safe Newton-Schulz init for SPD M)
__global__ __launch_bounds__(256) void k_scale_x0(const float* __restrict__ M,
                                                  float* __restrict__ X, const Ctrl* ctrl) {
    float nrm = __uint_as_float(*(volatile const unsigned*)&ctrl->norm_bits);
    float a = 1.0f / (nrm * nrm);
    size_t i = (size_t)blockIdx.x * 256 + threadIdx.x;
    X[i] = M[i] * a;
}

// c[j] = sum_k B1[k,j] * b[k]   (c = A^+ b, B1 = M^-1 A)
__global__ __launch_bounds__(256) void k_cvec(const float* __restrict__ B1,
                                              const float* __restrict__ b,
                                              float* __restrict__ c) {
    int j = blockIdx.x * 256 + threadIdx.x;
    float s = 0.0f;
    for (int k = 0; k < N1; ++k) s += B1[(size_t)k * N2 + j] * b[k];
    c[j] = s;
}

__device__ __forceinline__ float hash01(unsigned j) {
    unsigned h = j * 0x9E3779B9u + 0x85EBCA6Bu;
    h ^= h >> 16; h *= 0x85EBCA6Bu;
    h ^= h >> 13; h *= 0xC2B2AE35u;
    h ^= h >> 16;
    return (float)(h >> 8) * (1.0f / 16777216.0f);  // [0,1)
}

// z0: one uniform row of length N2 broadcast over batch
__global__ __launch_bounds__(256) void k_z0(float* __restrict__ Z) {
    size_t i = (size_t)blockIdx.x * 256 + threadIdx.x;
    Z[i] = hash01((unsigned)(i % N2));
}

// x = clip(z,0,1); u = 2x - z - alpha*w
__global__ __launch_bounds__(256) void k_pre(const float* __restrict__ Z,
                                             const float* __restrict__ W,
                                             float* __restrict__ U,
                                             float* __restrict__ X, const Ctrl* ctrl) {
    if (*(volatile const int*)&ctrl->flag) return;
    size_t i = (size_t)blockIdx.x * 256 + threadIdx.x;
    float z = Z[i];
    float x = fminf(fmaxf(z, 0.0f), 1.0f);
    X[i] = x;
    U[i] = 2.0f * x - z - ALPHA * W[i];
}

// z_new = z - x + y; accumulate ||y - x||^2
__global__ __launch_bounds__(256) void k_update(float* __restrict__ Z,
                                                const float* __restrict__ X,
                                                const float* __restrict__ Y, Ctrl* ctrl) {
    if (*(volatile const int*)&ctrl->flag) return;
    __shared__ float s[256];
    size_t i = (size_t)blockIdx.x * 256 + threadIdx.x;
    float y = Y[i], x = X[i], z = Z[i];
    Z[i] = z - x + y;
    float d = y - x;
    s[threadIdx.x] = d * d;
    __syncthreads();
    for (int st = 128; st > 0; st >>= 1) {
        if (threadIdx.x < st) s[threadIdx.x] += s[threadIdx.x + st];
        __syncthreads();
    }
    if (threadIdx.x == 0) atomicAdd(&ctrl->diff2, s[0]);
}

__global__ void k_check(Ctrl* c, const int* __restrict__ max_depth) {
    if (c->flag) return;
    int d = c->depth + 1;
    c->depth = d;
    float diff = sqrtf(c->diff2);
    if (diff <= EPSTOL || d >= *max_depth) c->flag = 1;
    c->diff2 = 0.0f;
}

__global__ __launch_bounds__(256) void k_final(const float* __restrict__ Z,
                                               float* __restrict__ out) {
    size_t i = (size_t)blockIdx.x * 256 + threadIdx.x;
    out[i] = fminf(fmaxf(Z[i], 0.0f), 1.0f);
}

// ---------------------------------------------------------------------------
extern "C" void kernel_launch(void* const* d_in, const int* in_sizes, int n_in,
                              void* d_out, int out_size, void* d_ws, size_t ws_size,
                              hipStream_t stream) {
    const float* dmat  = (const float*)d_in[0];   // [BATCH, DDIM]
    const float* A     = (const float*)d_in[1];   // [N1, N2]
    const float* bvec  = (const float*)d_in[2];   // [N1]
    const float* W_lin = (const float*)d_in[3];   // [DDIM, N2]
    const float* b_lin = (const float*)d_in[4];   // [N2]
    const int*   mdep  = (const int*)d_in[5];     // scalar max_depth
    float* out = (float*)d_out;                   // [BATCH, N2]

    char* ws = (char*)d_ws;
    Ctrl* ctrl = (Ctrl*)ws;
    float* base = (float*)(ws + 256);
    const size_t SZ_M  = (size_t)N1 * N1;
    const size_t SZ_BN = (size_t)N1 * N2;
    const size_t SZ_P  = (size_t)N2 * N2;
    const size_t SZ_BZ = (size_t)BATCH * N2;

    float* Mw = base;             // A A^T           [N1,N1]
    float* XA = Mw + SZ_M;        // Newton ping
    float* XB = XA + SZ_M;        // Newton pong
    float* Tw = XB + SZ_M;        // Newton temp
    float* B1 = Tw + SZ_M;        // M^-1 A          [N1,N2]
    float* Pw = B1 + SZ_BN;       // I - A^T M^-1 A  [N2,N2]
    float* cw = Pw + SZ_P;        // A^+ b           [N2]
    float* Ww = cw + N2;          // cost w          [BATCH,N2]
    float* Zw = Ww + SZ_BZ;       // iterate z
    float* Uw = Zw + SZ_BZ;       // 2x - z - a*w
    float* XC = Uw + SZ_BZ;       // clip(z)
    float* Yw = XC + SZ_BZ;       // project_C2(u)

    const int* zf = &ctrl->zero;  // never-skip flag for setup GEMMs
    const int* cf = &ctrl->flag;  // convergence flag for iteration GEMMs
    dim3 blk(256);
    dim3 g512(512 / 128, 512 / 64);     // (4, 8)
    dim3 gBxN2(N2 / 128, BATCH / 64);   // (16, 8)
    dim3 gN2(N2 / 128, N2 / 64);        // (16, 32)
    const int nEW = (int)(SZ_BZ / 256); // elementwise blocks (4096)

    k_init<<<1, 1, 0, stream>>>(ctrl);

    // M = A A^T
    k_gemm<false, true, 0><<<g512, blk, 0, stream>>>(A, A, nullptr, Mw,
                                                     N2, N2, N2, N1, zf);
    // ||M||_1 -> X0 = M / ||M||_1^2
    k_rownorm<<<N1, blk, 0, stream>>>(Mw, ctrl);
    k_scale_x0<<<(int)(SZ_M / 256), blk, 0, stream>>>(Mw, XA, ctrl);

    // Newton-Schulz: X <- X(2I - M X)
    float* Xc = XA;
    float* Xn = XB;
    for (int it = 0; it < NEWTON_ITERS; ++it) {
        k_gemm<false, false, 0><<<g512, blk, 0, stream>>>(Mw, Xc, nullptr, Tw,
                                                          N1, N1, N1, N1, zf);
        k_gemm<false, false, 2><<<g512, blk, 0, stream>>>(Xc, Tw, Xc, Xn,
                                                          N1, N1, N1, N1, zf);
        float* t = Xc; Xc = Xn; Xn = t;
    }
    float* Minv = Xc;

    // B1 = M^-1 A ;  P = I - B1^T A ;  c = B1^T b
    k_gemm<false, false, 0><<<gBxN2, blk, 0, stream>>>(Minv, A, nullptr, B1,
                                                       N1, N1, N2, N2, zf);
    k_gemm<true, false, 3><<<gN2, blk, 0, stream>>>(B1, A, nullptr, Pw,
                                                    N1, N2, N2, N2, zf);
    k_cvec<<<N2 / 256, blk, 0, stream>>>(B1, bvec, cw);

    // w = d @ W_lin + b_lin
    k_gemm<false, false, 1><<<gBxN2, blk, 0, stream>>>(dmat, W_lin, b_lin, Ww,
                                                       DDIM, DDIM, N2, N2, zf);
    // z0
    k_z0<<<nEW, blk, 0, stream>>>(Zw);

    // DYS iterations (flag-gated => emulates reference while-loop)
    for (int it = 0; it < DYS_ITERS; ++it) {
        k_pre<<<nEW, blk, 0, stream>>>(Zw, Ww, Uw, XC, ctrl);
        k_gemm<false, false, 1><<<gBxN2, blk, 0, stream>>>(Uw, Pw, cw, Yw,
                                                           N2, N2, N2, N2, cf);
        k_update<<<nEW, blk, 0, stream>>>(Zw, XC, Yw, ctrl);
        k_check<<<1, 1, 0, stream>>>(ctrl, mdep);
    }

    // out = clip(z, 0, 1)
    k_final<<<nEW, blk, 0, stream>>>(Zw, out);
}